// CortexModel_67035849555989
// MI455X (gfx1250) — compile-verified
//
#include <hip/hip_runtime.h>
#include <stdint.h>

// CDNA5 / gfx1250: wave32. Per 16-pixel group, out[p] = diag(W x P^T)[p],
// computed with V_WMMA_F32_16X16X4_F32 (fp32 in/out). mask * (+-gamma), padded
// with zeros to a multiple-of-4 row width, is precomputed into d_ws so the hot
// loops have no lane-divergent guards (EXEC stays all-ones around WMMA).
// Activation slab staged via GLOBAL_LOAD_ASYNC_TO_LDS_B32 (ASYNCcnt).
// K-reduction is split across wave pairs (u-rows) for 2x wave concurrency;
// partials combine in LDS during diagonal extraction. Wave-uniform loop
// bounds go through readfirstlane so loop control stays SALU (no exec masks).

typedef float v2f __attribute__((ext_vector_type(2)));
typedef float v8f __attribute__((ext_vector_type(8)));

namespace {
constexpr int XD   = 128;          // cortex X
constexpr int YD   = 128;          // cortex Y
constexpr int RFW  = 24;           // afferent RF width
constexpr int ERAD = 19;
constexpr int IRAD = 47;
constexpr int EWW  = 2 * ERAD + 1; // 39
constexpr int IWW  = 2 * IRAD + 1; // 95
constexpr int PADW = 47;
constexpr int IMGW = 152;
constexpr float GAM_E = 0.9f;
constexpr float GAM_I = 0.9f;

constexpr int YB    = 64;            // y-pixels per block
constexpr int LPW   = YB + IWW;      // 159 LDS cols (max used index 158)
constexpr int LROWS = IWW;           // 95 LDS rows
constexpr int NW    = 8;             // waves per block -> 256 threads
                                     // (wave pairs share a 16-pixel group)

// d_ws layout (floats): gamma-scaled, zero-padded masks.
constexpr int AFF_P  = 24;                      // 24 % 4 == 0, no pad
constexpr int EX_P   = 40;                      // 39 -> 40
constexpr int IN_P   = 96;                      // 95 -> 96
constexpr int WS_AFF = 0;
constexpr int WS_EX  = WS_AFF + RFW * AFF_P;    // 576
constexpr int WS_IN  = WS_EX + EWW * EX_P;      // 2136
constexpr int WS_TOT = WS_IN + IWW * IN_P;      // 11256 floats (~45 KB)
}

// ---------------------------------------------------------------------------
// Prep: ws[k] = mask * (+-gamma), rows padded with zeros.
// ---------------------------------------------------------------------------
__global__ void prep_masks(const float* __restrict__ affM,
                           const float* __restrict__ exM,
                           const float* __restrict__ inM,
                           float* __restrict__ ws)
{
  const int i = blockIdx.x * blockDim.x + threadIdx.x;
  if (i >= WS_TOT) return;
  float val;
  if (i < WS_EX) {
    val = affM[i];                                   // gamma = +1
  } else if (i < WS_IN) {
    const int k = i - WS_EX, u = k / EX_P, v = k % EX_P;
    val = (v < EWW) ? GAM_E * exM[u * EWW + v] : 0.0f;
  } else {
    const int k = i - WS_IN, u = k / IN_P, v = k % IN_P;
    val = (v < IWW) ? -GAM_I * inM[u * IWW + v] : 0.0f;
  }
  ws[i] = val;
}

// ---------------------------------------------------------------------------
// One lateral term over u-rows [u0,u1) (wave-uniform SGPR bounds).
// Compile-time KW/KP/OFF -> chunk loop unrolls x4 for load pipelining.
// ---------------------------------------------------------------------------
template<int KW, int KP, int OFF>
__device__ __forceinline__ v8f dot16_lateral(
    v8f acc, const float* __restrict__ W, const float* __restrict__ meff,
    const float* lds, int u0, int u1,
    int pix, int yl, int n, int hl)
{
  constexpr int  NFULL = KW >> 2;
  constexpr bool TAIL  = (KW & 3) != 0;
  const float* wbase = W + (size_t)pix * KW * KW;
  #pragma unroll 1
  for (int u = u0; u < u1; ++u) {
    const float* wrow = wbase + u * KW;
    const float* mrow = meff + u * KP;
    const float* prow = lds + (OFF + u) * LPW + (yl + n + OFF);
    __builtin_prefetch(wrow + KW, 0, 3);     // global_prefetch_b8: next row
    #pragma unroll 4
    for (int vb = 0; vb < NFULL; ++vb) {
      const int v = vb * 4 + 2 * hl;         // this lane's K pair: v, v+1
      v2f w; w.x = wrow[v]; w.y = wrow[v + 1];
      const v2f m = *(const v2f*)(mrow + v);       // 8B-aligned b64
      v2f b; b.x = prow[v]; b.y = prow[v + 1];     // LDS patch (read2)
      const v2f a = w * m;                         // v_pk_mul_f32
      acc = __builtin_amdgcn_wmma_f32_16x16x4_f32(false, a, false, b,
                                                  (short)0, acc, false, false);
    }
    if constexpr (TAIL) {
      const int v  = NFULL * 4 + 2 * hl;
      const int i1 = (v + 1 < KW) ? (v + 1) : (KW - 1);  // clamp: no OOB read
      v2f w; w.x = wrow[v]; w.y = wrow[i1];
      const v2f m = *(const v2f*)(mrow + v);   // m.y == 0 for invalid k
      v2f b; b.x = prow[v]; b.y = prow[v + 1]; // slab padded: finite*0
      const v2f a = w * m;
      acc = __builtin_amdgcn_wmma_f32_16x16x4_f32(false, a, false, b,
                                                  (short)0, acc, false, false);
    }
  }
  return acc;
}

// Afferent term over u-rows [u0,u1): gathered 24x24 window from the input.
__device__ __forceinline__ v8f dot16_afferent(
    v8f acc, const float* __restrict__ W, const float* __restrict__ meff,
    const float* __restrict__ img, const int* __restrict__ rfs,
    int u0, int u1, int pix, int hl)
{
  const int rs = rfs[2 * pix + 0];
  const int cs = rfs[2 * pix + 1];
  const float* wbase = W + (size_t)pix * (RFW * RFW);
  #pragma unroll 1
  for (int u = u0; u < u1; ++u) {
    const float* wrow = wbase + u * RFW;
    const float* mrow = meff + u * AFF_P;
    const float* irow = img + (rs + u) * IMGW + cs;
    #pragma unroll 3
    for (int vb = 0; vb < RFW / 4; ++vb) {
      const int v = vb * 4 + 2 * hl;
      const v2f w = *(const v2f*)(wrow + v);   // pix*576+u*24+v even: aligned
      const v2f m = *(const v2f*)(mrow + v);
      v2f b; b.x = irow[v]; b.y = irow[v + 1]; // cs parity unknown: b32 pair
      const v2f a = w * m;
      acc = __builtin_amdgcn_wmma_f32_16x16x4_f32(false, a, false, b,
                                                  (short)0, acc, false, false);
    }
  }
  return acc;
}

__launch_bounds__(NW * 32)
__global__ void cortex_wmma_kernel(
    const float* __restrict__ input,      // (1,152,152)
    const float* __restrict__ activation, // (1,128,128)
    const float* __restrict__ affW,       // (1,128,128,24,24)
    const float* __restrict__ exW,        // (1,128,128,39,39)
    const float* __restrict__ inW,        // (1,128,128,95,95)
    const int*   __restrict__ rfs,        // (128,128,2)
    const float* __restrict__ ws,         // padded gamma-scaled masks
    float* __restrict__ out)              // (1,128,128)
{
  __shared__ float lds_act[LROWS * LPW];  // ~60.4 KB slab, reused for combine

  const int x   = blockIdx.x;
  const int y0b = blockIdx.y * YB;

  // Stage padded activation rows [x, x+94] x padded cols [y0b, y0b+158].
  // Interior cells: async DMA global->LDS (ASYNCcnt). Border/pad: zero stores.
  {
    const uint64_t abase = (uint64_t)(uintptr_t)activation;
    for (int i = threadIdx.x; i < LROWS * LPW; i += blockDim.x) {
      const int u  = i / LPW;
      const int c  = i % LPW;
      const int ar = x + u - PADW;
      const int ac = y0b + c - PADW;
      if (ar >= 0 && ar < XD && ac >= 0 && ac < YD) {
        const unsigned ldsoff = (unsigned)(uintptr_t)(&lds_act[i]);
        const uint64_t ga = abase + 4ull * (unsigned)(ar * YD + ac);
        asm volatile("global_load_async_to_lds_b32 %0, %1, off"
                     :: "v"(ldsoff), "v"(ga) : "memory");
      } else {
        lds_act[i] = 0.0f;
      }
    }
    asm volatile("s_wait_asynccnt 0x0" ::: "memory");
  }
  __syncthreads();

  const int lane = threadIdx.x & 31;
  // Wave index is wave-uniform: pin it to an SGPR so loop bounds derived from
  // it are scalar -> pure s_cmp/s_cbranch loops, no exec masking near WMMA.
  const int wv   = __builtin_amdgcn_readfirstlane(threadIdx.x >> 5); // 0..7
  const int grp  = wv >> 1;             // pixel group 0..3
  const int us   = wv & 1;              // u-range split within the pair
  const int hl   = lane >> 4;           // selects K={0,1} vs K={2,3}
  const int n    = lane & 15;           // pixel-in-group (A row m == B col n)
  const int yl   = grp * 16;
  const int pix  = x * YD + (y0b + yl + n);

  v8f acc = {0.f, 0.f, 0.f, 0.f, 0.f, 0.f, 0.f, 0.f};

  // Each wave of a pair does half the u-rows of every term.
  acc = dot16_afferent(acc, affW, ws + WS_AFF, input, rfs,
                       us * (RFW / 2), (us + 1) * (RFW / 2), pix, hl);
  acc = dot16_lateral<EWW, EX_P, PADW - ERAD>(acc, exW, ws + WS_EX, lds_act,
                       us * 20, us ? EWW : 20, pix, yl, n, hl);
  acc = dot16_lateral<IWW, IN_P, PADW - IRAD>(acc, inW, ws + WS_IN, lds_act,
                       us * 48, us ? IWW : 48, pix, yl, n, hl);

  // Combine wave-pair partials and extract diag(D): D[m][n] lives in VGPR m%8,
  // lane n + 16*(m/8). Slab is dead now -> reuse as scratch.
  __syncthreads();
  float* dscr = lds_act;
  {
    float* myrow = dscr + (wv * 32 + lane) * 8;
    #pragma unroll
    for (int r = 0; r < 8; ++r) myrow[r] = acc[r];
  }
  __syncthreads();   // cross-wave combine below
  if (us == 0 && lane < 16) {
    const int m       = lane;
    const int srcLane = (m < 8) ? m : (m + 16);
    const int e       = srcLane * 8 + (m & 7);
    const float v0 = dscr[(wv * 32) * 8 + e];
    const float v1 = dscr[((wv + 1) * 32) * 8 + e];
    out[x * YD + y0b + yl + m] = v0 + v1;
  }
}

extern "C" void kernel_launch(void* const* d_in, const int* in_sizes, int n_in,
                              void* d_out, int out_size, void* d_ws, size_t ws_size,
                              hipStream_t stream) {
  const float* input = (const float*)d_in[0];
  const float* act   = (const float*)d_in[1];
  const float* affW  = (const float*)d_in[2];
  const float* exW   = (const float*)d_in[3];
  const float* inW   = (const float*)d_in[4];
  const float* affM  = (const float*)d_in[5];
  const float* exM   = (const float*)d_in[6];
  const float* inM   = (const float*)d_in[7];
  const int*   rfs   = (const int*)d_in[8];
  float* ws  = (float*)d_ws;   // needs WS_TOT*4 ~= 45 KB of scratch
  float* out = (float*)d_out;

  hipLaunchKernelGGL(prep_masks, dim3((WS_TOT + 255) / 256), dim3(256), 0,
                     stream, affM, exM, inM, ws);

  dim3 grid(XD, YD / YB);   // 128 x 2 blocks
  dim3 block(NW * 32);      // 256 threads = 8 waves (wave32)
  hipLaunchKernelGGL(cortex_wmma_kernel, grid, block, 0, stream,
                     input, act, affW, exW, inW, rfs, ws, out);
}